// GNN_31671088841437
// MI455X (gfx1250) — compile-verified
//
#include <hip/hip_runtime.h>
#include <hip/hip_bf16.h>
#include <stdint.h>

// ---------------- problem constants (match the reference) ----------------
#define NN      50000      // nodes
#define NE      600000     // edges
#define DD      128        // feature dim
#define RR      8          // relations
#define MAXDEG  10
#define NGRAPH  64
#define NBLOCKS 2
#define NTILES  ((NN + 15) / 16)   // 3125 row tiles of 16

typedef __attribute__((ext_vector_type(2))) float v2f;
typedef __attribute__((ext_vector_type(8))) float v8f;

// =======================================================================
// Edge setup: etype = argmax(edge_attr), per-(rel,dst) counts, in-degree
// =======================================================================
__global__ __launch_bounds__(256) void edge_setup_k(
    const float* __restrict__ eattr, const int* __restrict__ dst,
    int* __restrict__ etype, int* __restrict__ rcnt, int* __restrict__ totcnt)
{
    int e = blockIdx.x * 256 + threadIdx.x;
    if (e >= NE) return;
    const float* a = eattr + (size_t)e * RR;
    int best = 0; float bv = a[0];
#pragma unroll
    for (int r = 1; r < RR; ++r) { float v = a[r]; if (v > bv) { bv = v; best = r; } }
    etype[e] = best;
    int d = dst[e];
    atomicAdd(&rcnt[best * NN + d], 1);
    atomicAdd(&totcnt[d], 1);
}

// ---------------- degree grouping: hist -> scan -> fill ----------------
__global__ __launch_bounds__(256) void deg_hist_k(const int* __restrict__ totcnt,
                                                  int* __restrict__ hist)
{
    int i = blockIdx.x * 256 + threadIdx.x;
    if (i >= NN) return;
    int c = totcnt[i]; int k = (c < MAXDEG) ? c : MAXDEG;
    atomicAdd(&hist[k], 1);
}

__global__ void deg_scan_k(const int* __restrict__ hist, int* __restrict__ off,
                           int* __restrict__ cur)
{
    if (threadIdx.x == 0 && blockIdx.x == 0) {
        int acc = 0;
        for (int k = 0; k <= MAXDEG; ++k) { off[k] = acc; cur[k] = acc; acc += hist[k]; }
        off[MAXDEG + 1] = acc;
    }
}

__global__ __launch_bounds__(256) void deg_fill_k(const int* __restrict__ totcnt,
                                                  int* __restrict__ cur,
                                                  int* __restrict__ deglist)
{
    int i = blockIdx.x * 256 + threadIdx.x;
    if (i >= NN) return;
    int c = totcnt[i]; int k = (c < MAXDEG) ? c : MAXDEG;
    int pos = atomicAdd(&cur[k], 1);
    deglist[pos] = i;
}

// =======================================================================
// Elementwise ReLU (in place), float4-vectorized
// =======================================================================
__global__ __launch_bounds__(256) void relu_k(float* __restrict__ h, int n4)
{
    int i = blockIdx.x * 256 + threadIdx.x;
    if (i >= n4) return;
    float4* p = (float4*)h;
    float4 v = p[i];
    v.x = fmaxf(v.x, 0.f); v.y = fmaxf(v.y, 0.f);
    v.z = fmaxf(v.z, 0.f); v.w = fmaxf(v.w, 0.f);
    p[i] = v;
}

// =======================================================================
// Edge scatter: agg[dst] += h[src]  (rel >= 0 filters on etype == rel)
// 32 lanes per edge, float4 gather + fp32 global atomics
// =======================================================================
__global__ __launch_bounds__(256) void scatter_k(
    const float* __restrict__ h, const int* __restrict__ src,
    const int* __restrict__ dst, const int* __restrict__ etype, int rel,
    float* __restrict__ agg)
{
    long tid = (long)blockIdx.x * 256 + threadIdx.x;
    int e = (int)(tid >> 5);
    int lane = (int)(tid & 31);
    if (e >= NE) return;
    if (rel >= 0 && etype[e] != rel) return;
    int s = src[e], d = dst[e];
    const float4* hs = (const float4*)(h + (size_t)s * DD);
    float4 v = hs[lane];                       // 32 lanes * 4 = 128 floats
    float* ad = agg + (size_t)d * DD + lane * 4;
    atomicAdd(ad + 0, v.x); atomicAdd(ad + 1, v.y);
    atomicAdd(ad + 2, v.z); atomicAdd(ad + 3, v.w);
}

// =======================================================================
// WMMA fp32 GEMM: C[rows] (op)= A[rows] @ B (+bias) (relu?)
//   - B is DD x DD row-major (in,out), resident in L2 (64 KB)
//   - one wave -> one 16x16 output tile, K=128 via 32x v_wmma_f32_16x16x4_f32
//   - 8 waves / 256-thread block cover a 16x128 strip; A strip staged in LDS
//   - SCALE: A row scaled by 1/max(cnt[row],1)   (segment-mean fusion)
//   - GATHER: rows taken from rowidx[segoff[seg] .. segoff[seg+1])
// =======================================================================
template<bool SCALE, bool GATHER, bool ACCUM, bool BIAS, bool RELUOUT>
__global__ __launch_bounds__(256) void gemm16_k(
    const float* __restrict__ A, const float* __restrict__ B,
    const float* __restrict__ bias, float* __restrict__ C,
    const int* __restrict__ rowidx, const int* __restrict__ segoff, int seg,
    const int* __restrict__ cnt, int nrows_const)
{
    __shared__ float As[16][132];   // pitch 132: 16-lane column reads hit distinct banks

    int start = 0, nrows = nrows_const;
    if (GATHER) { start = segoff[seg]; nrows = segoff[seg + 1] - start; }
    int row0 = blockIdx.x * 16;
    if (row0 >= nrows) return;      // block-uniform exit (EXEC stays all-ones for WMMA)

    int t = threadIdx.x;

    // ---- stage A strip (16 x 128) into LDS; fuse gather / mean-scale ----
    {
        int r  = t >> 4;            // 0..15  (strip row)
        int c0 = (t & 15) * 8;      // 8 consecutive floats per thread
        int gr = row0 + r;
        float vals[8];
        if (gr < nrows) {
            int arow = GATHER ? rowidx[start + gr] : gr;
            const float* Ar = A + (size_t)arow * DD + c0;
            float scale = 1.f;
            if (SCALE) { int c = cnt[arow]; scale = 1.f / (float)((c > 1) ? c : 1); }
#pragma unroll
            for (int i = 0; i < 8; ++i) vals[i] = Ar[i] * scale;
        } else {
#pragma unroll
            for (int i = 0; i < 8; ++i) vals[i] = 0.f;
        }
#pragma unroll
        for (int i = 0; i < 8; ++i) As[r][c0 + i] = vals[i];
    }
    __syncthreads();

    // ---- per-wave 16x16 tile ----
    int wave = t >> 5;
    int lane = t & 31;
    int n0   = wave * 16;           // output column tile
    int m    = lane & 15;           // A row (lanes 0-15) / row (lanes 16-31)
    int koff = (lane >> 4) << 1;    // K sub-slot: lanes 16-31 hold K+2,K+3

    const float* Bp = B + n0 + m;   // B[k][n0+m] column walker
    v8f c = {};
#pragma unroll
    for (int kt = 0; kt < DD; kt += 4) {
        int k = kt + koff;
        v2f a, b;
        a.x = As[m][k];
        a.y = As[m][k + 1];
        b.x = Bp[(size_t)k * DD];
        b.y = Bp[(size_t)(k + 1) * DD];
        // D = A(16x4) x B(4x16) + C ; full fp32, VOP3P WMMA
        c = __builtin_amdgcn_wmma_f32_16x16x4_f32(
                false, a, false, b, (short)0, c, false, false);
    }

    // ---- epilogue: bias / accumulate / relu, honoring C/D VGPR layout ----
    float bv = 0.f;
    if (BIAS) bv = bias[n0 + m];
#pragma unroll
    for (int i = 0; i < 8; ++i) {
        int mr = i + ((lane >> 4) << 3);   // lanes 0-15 -> M=i ; 16-31 -> M=i+8
        int gr = row0 + mr;
        if (gr < nrows) {
            int crow = GATHER ? rowidx[start + gr] : gr;
            float v = c[i];
            if (BIAS)  v += bv;
            size_t cidx = (size_t)crow * DD + n0 + m;
            if (ACCUM) v += C[cidx];
            if (RELUOUT) v = fmaxf(v, 0.f);
            C[cidx] = v;
        }
    }
}

// =======================================================================
// lin2 (128 -> 1) + graph pooling: out[batch[i]] += h_i . w + b
// one wave per node, shuffle reduction
// =======================================================================
__global__ __launch_bounds__(256) void lin2_pool_k(
    const float* __restrict__ h, const float* __restrict__ w,
    const float* __restrict__ b, const int* __restrict__ batch,
    float* __restrict__ out)
{
    int node = blockIdx.x * 8 + (threadIdx.x >> 5);
    int lane = threadIdx.x & 31;
    if (node >= NN) return;
    const float4* hr = (const float4*)(h + (size_t)node * DD);
    const float4* wr = (const float4*)w;
    float4 hv = hr[lane], wv = wr[lane];
    float v = hv.x * wv.x + hv.y * wv.y + hv.z * wv.z + hv.w * wv.w;
#pragma unroll
    for (int off = 16; off > 0; off >>= 1) v += __shfl_down(v, off, 32);
    if (lane == 0) atomicAdd(&out[batch[node]], v + b[0]);
}

// =======================================================================
// host-side orchestration
// =======================================================================
extern "C" void kernel_launch(void* const* d_in, const int* in_sizes, int n_in,
                              void* d_out, int out_size, void* d_ws, size_t ws_size,
                              hipStream_t stream)
{
    (void)in_sizes; (void)n_in; (void)out_size; (void)ws_size;

    const float* x       = (const float*)d_in[0];
    const int*   eidx    = (const int*)  d_in[1];
    const float* eattr   = (const float*)d_in[2];
    const int*   batch   = (const int*)  d_in[3];
    const float* W_emb   = (const float*)d_in[4];
    const float* rgcn_w  = (const float*)d_in[5];   // [NB][R][D][D]
    const float* rgcn_rt = (const float*)d_in[6];   // [NB][D][D]
    const float* rgcn_b  = (const float*)d_in[7];   // [NB][D]
    const float* mf_wl   = (const float*)d_in[8];   // [NB][11][D][D]
    const float* mf_bl   = (const float*)d_in[9];   // [NB][11][D]
    const float* mf_wr   = (const float*)d_in[10];  // [NB][11][D][D]
    const float* gc_wrel = (const float*)d_in[11];
    const float* gc_brel = (const float*)d_in[12];
    const float* gc_wrt  = (const float*)d_in[13];
    const float* lin1_w  = (const float*)d_in[14];
    const float* lin1_b  = (const float*)d_in[15];
    const float* lin2_w  = (const float*)d_in[16];
    const float* lin2_b  = (const float*)d_in[17];

    const int* src = eidx;
    const int* dst = eidx + NE;

    // ---- carve workspace (256B-aligned slots) ----
    uintptr_t base = (uintptr_t)d_ws;
    size_t cur = 0;
    auto take = [&](size_t bytes) -> void* {
        void* p = (void*)(base + cur);
        cur += (bytes + 255) & ~(size_t)255;
        return p;
    };
    float* h      = (float*)take((size_t)NN * DD * 4);
    float* h2     = (float*)take((size_t)NN * DD * 4);
    float* agg    = (float*)take((size_t)NN * DD * 4);
    int*   etype  = (int*)  take((size_t)NE * 4);
    int*   rcnt   = (int*)  take((size_t)RR * NN * 4);
    int*   totcnt = (int*)  take((size_t)NN * 4);
    int*   deglist= (int*)  take((size_t)NN * 4);
    int*   smalls = (int*)  take(64 * 4);
    int* hist = smalls;         // 11
    int* doff = smalls + 16;    // 12
    int* dcur = smalls + 32;    // 11

    const int GEB = 256;
    const int egrid   = (NE + GEB - 1) / GEB;
    const int ngrid   = (NN + GEB - 1) / GEB;
    const int sgrid   = (int)(((long)NE * 32 + GEB - 1) / GEB);
    const int r4grid  = (NN * DD / 4 + GEB - 1) / GEB;

    // ---- graph structure (counts, etype, degree lists) ----
    hipMemsetAsync(rcnt,   0, (size_t)RR * NN * 4, stream);
    hipMemsetAsync(totcnt, 0, (size_t)NN * 4,      stream);
    hipMemsetAsync(smalls, 0, 64 * 4,              stream);
    edge_setup_k<<<egrid, GEB, 0, stream>>>(eattr, dst, etype, rcnt, totcnt);
    deg_hist_k<<<ngrid, GEB, 0, stream>>>(totcnt, hist);
    deg_scan_k<<<1, 32, 0, stream>>>(hist, doff, dcur);
    deg_fill_k<<<ngrid, GEB, 0, stream>>>(totcnt, dcur, deglist);

    // ---- node embedding: h = x @ W_emb ----
    gemm16_k<false,false,false,false,false><<<NTILES, 256, 0, stream>>>(
        x, W_emb, nullptr, h, nullptr, nullptr, 0, nullptr, NN);

    // ---- convolution blocks ----
    for (int b = 0; b < NBLOCKS; ++b) {
        // h = relu(h)
        relu_k<<<r4grid, GEB, 0, stream>>>(h, NN * DD / 4);

        // RGCN: h2 = h @ root + bias
        gemm16_k<false,false,false,true,false><<<NTILES, 256, 0, stream>>>(
            h, rgcn_rt + (size_t)b * DD * DD, rgcn_b + (size_t)b * DD,
            h2, nullptr, nullptr, 0, nullptr, NN);
        //        h2 += mean_r(agg) @ W_r  for each relation
        for (int r = 0; r < RR; ++r) {
            hipMemsetAsync(agg, 0, (size_t)NN * DD * 4, stream);
            scatter_k<<<sgrid, GEB, 0, stream>>>(h, src, dst, etype, r, agg);
            gemm16_k<true,false,true,false,false><<<NTILES, 256, 0, stream>>>(
                agg, rgcn_w + ((size_t)b * RR + r) * DD * DD, nullptr,
                h2, nullptr, nullptr, 0, rcnt + (size_t)r * NN, NN);
        }

        // h2 = relu(h2)
        relu_k<<<r4grid, GEB, 0, stream>>>(h2, NN * DD / 4);

        // MFConv: agg = segment_sum(h2[src], dst)
        hipMemsetAsync(agg, 0, (size_t)NN * DD * 4, stream);
        scatter_k<<<sgrid, GEB, 0, stream>>>(h2, src, dst, etype, -1, agg);
        // per-degree-group GEMMs: h = agg@Wl[k] + bl[k] + h2@Wr[k]
        for (int k = 0; k <= MAXDEG; ++k) {
            gemm16_k<false,true,false,true,false><<<NTILES, 256, 0, stream>>>(
                agg, mf_wl + ((size_t)b * (MAXDEG + 1) + k) * DD * DD,
                mf_bl + ((size_t)b * (MAXDEG + 1) + k) * DD,
                h, deglist, doff, k, nullptr, 0);
            gemm16_k<false,true,true,false,false><<<NTILES, 256, 0, stream>>>(
                h2, mf_wr + ((size_t)b * (MAXDEG + 1) + k) * DD * DD, nullptr,
                h, deglist, doff, k, nullptr, 0);
        }
    }

    // ---- GraphConv(mean): h2 = mean_agg(h) @ wrel + brel + h @ wroot ----
    hipMemsetAsync(agg, 0, (size_t)NN * DD * 4, stream);
    scatter_k<<<sgrid, GEB, 0, stream>>>(h, src, dst, etype, -1, agg);
    gemm16_k<true,false,false,true,false><<<NTILES, 256, 0, stream>>>(
        agg, gc_wrel, gc_brel, h2, nullptr, nullptr, 0, totcnt, NN);
    gemm16_k<false,false,true,false,false><<<NTILES, 256, 0, stream>>>(
        h, gc_wrt, nullptr, h2, nullptr, nullptr, 0, nullptr, NN);

    // ---- lin1 + relu: h = relu(h2 @ lin1_w + lin1_b) ----
    gemm16_k<false,false,false,true,true><<<NTILES, 256, 0, stream>>>(
        h2, lin1_w, lin1_b, h, nullptr, nullptr, 0, nullptr, NN);

    // ---- lin2 + graph pooling ----
    hipMemsetAsync(d_out, 0, NGRAPH * 4, stream);
    lin2_pool_k<<<(NN + 7) / 8, GEB, 0, stream>>>(h, lin2_w, lin2_b, batch,
                                                  (float*)d_out);
}